// KGTransformer_48610439856748
// MI455X (gfx1250) — compile-verified
//
#include <hip/hip_runtime.h>
#include <hip/hip_bf16.h>
#include <math.h>

// ---------------------------------------------------------------------------
// KG-Transformer (HGT-style) for gfx1250 / MI455X.
// fp32 path using V_WMMA_F32_16X16X4_F32 (CDNA5 fp32 matrix op), LDS-staged
// A-tiles via GLOBAL_LOAD_ASYNC_TO_LDS_B128 (ASYNCcnt-tracked).
// ---------------------------------------------------------------------------

typedef __attribute__((ext_vector_type(2))) float v2f;
typedef __attribute__((ext_vector_type(8))) float v8f;
typedef __attribute__((ext_vector_type(4))) int   v4i;

#define NN   20000
#define EE   320000
#define DINN 128
#define HIDD 128
#define HH   8
#define HS   16
#define NTT  16
#define RTT  32
#define LLL  2
#define OUTD 128

#define LDA  132   // padded LDS row stride (floats): 528B = 16B-aligned, bank-skewed

#if defined(__has_builtin)
#if __has_builtin(__builtin_amdgcn_global_load_async_to_lds_b128) && \
    __has_builtin(__builtin_amdgcn_s_wait_asynccnt)
#define USE_ASYNC_LDS 1
#endif
#endif
#ifndef USE_ASYNC_LDS
#define USE_ASYNC_LDS 0
#endif

#define AS1 __attribute__((address_space(1)))
#define AS3 __attribute__((address_space(3)))

__device__ __forceinline__ void stage16(const float* gp, float* lp) {
#if USE_ASYNC_LDS
  // GLOBAL_LOAD_ASYNC_TO_LDS_B128: memory -> LDS, tracked by ASYNCcnt
  __builtin_amdgcn_global_load_async_to_lds_b128(
      (AS1 v4i*)gp, (AS3 v4i*)lp, /*imm_offset=*/0, /*cpol=*/0);
#else
  *(float4*)lp = *(const float4*)gp;
#endif
}

__device__ __forceinline__ void stage_fence() {
#if USE_ASYNC_LDS
  __builtin_amdgcn_s_wait_asynccnt(0);
#endif
  __syncthreads();
}

__device__ __forceinline__ v8f wmma_f32_k4(v2f a, v2f b, v8f c) {
  // D = A(16x4,f32) * B(4x16,f32) + C(16x16,f32)
  return __builtin_amdgcn_wmma_f32_16x16x4_f32(
      /*neg_a=*/false, a, /*neg_b=*/false, b,
      /*c_mod=*/(short)0, c, /*reuse_a=*/false, /*reuse_b=*/false);
}

__device__ __forceinline__ void atomicMaxF(float* addr, float v) {
  // Exact IEEE float max via monotone int mapping (init value must be -inf).
  if (v >= 0.0f) atomicMax((int*)addr, __float_as_int(v));
  else           atomicMin((unsigned int*)addr, __float_as_uint(v));
}

// ------------------------------ bucketing ----------------------------------

__global__ void init_counts_kernel(int* cnt_nt, int* cnt_rt) {
  int i = threadIdx.x;
  if (i < NTT) cnt_nt[i] = 0;
  if (i < RTT) cnt_rt[i] = 0;
}

__global__ void bucket_nodes_kernel(const int* __restrict__ ntype,
                                    int* cnt_nt, int* nodelist) {
  int n = blockIdx.x * blockDim.x + threadIdx.x;
  if (n >= NN) return;
  int t = ntype[n];
  int pos = atomicAdd(&cnt_nt[t], 1);
  nodelist[t * NN + pos] = n;
}

__global__ void bucket_edges_kernel(const int* __restrict__ etype,
                                    int* cnt_rt, int* edgelist) {
  int e = blockIdx.x * blockDim.x + threadIdx.x;
  if (e >= EE) return;
  int t = etype[e];
  int pos = atomicAdd(&cnt_rt[t], 1);
  edgelist[t * EE + pos] = e;
}

__global__ void reset_kernel(float* hag, float* amax, float* asum) {
  int i = blockIdx.x * blockDim.x + threadIdx.x;
  if (i < NN * HIDD) hag[i] = 0.0f;
  if (i < NN * HH) { asum[i] = 0.0f; amax[i] = -INFINITY; }
}

// ------------------- typed K/Q/V projection (WMMA GEMM) --------------------
// grid = (NN/16, NT), block = 256 (8 waves; wave w owns 16-column tile w).
// A tile (16 gathered rows x 128) staged once per block into LDS.

__global__ __launch_bounds__(256) void kqv_kernel(
    const float* __restrict__ xin,
    const float* __restrict__ WkL, const float* __restrict__ WqL,
    const float* __restrict__ WvL,
    const int* __restrict__ cnt_nt, const int* __restrict__ nodelist,
    float* __restrict__ kb, float* __restrict__ qb, float* __restrict__ vb) {
  const int t = blockIdx.y;
  const int cnt = cnt_nt[t];
  const int tile = blockIdx.x;
  if (tile * 16 >= cnt) return;

  __shared__ int sid[16];   // node id (clamped for safe loads)
  __shared__ int sok[16];   // original node id or -1 if tail-invalid
  __shared__ __align__(16) float sA[16 * LDA];
  if (threadIdx.x < 16) {
    int idx = tile * 16 + threadIdx.x;
    int n = (idx < cnt) ? nodelist[t * NN + idx] : -1;
    sok[threadIdx.x] = n;
    sid[threadIdx.x] = (n < 0) ? 0 : n;
  }
  __syncthreads();

  // stage 16x128 A tile into LDS (each thread moves 2x16B)
  {
    const int tid = threadIdx.x;
#pragma unroll
    for (int it = 0; it < 2; ++it) {
      int qd = it * 256 + tid;        // 512 x 16B chunks total
      int row = qd >> 5;              // 32 chunks per row
      int c4 = (qd & 31) * 4;
      stage16(xin + (size_t)sid[row] * DINN + c4, &sA[row * LDA + c4]);
    }
    stage_fence();
  }

  const int lane = threadIdx.x & 31;
  const int wt = threadIdx.x >> 5;           // column tile 0..7
  const int rc = lane & 15;                  // row (A) / col (B,D) index
  const int koff = (lane < 16) ? 0 : 2;      // K offset per ISA layout
  const bool full = (tile * 16 + 16) <= cnt;

  int rowid[8]; int okid[8];
#pragma unroll
  for (int i = 0; i < 8; ++i) {
    int m = i + ((lane < 16) ? 0 : 8);
    rowid[i] = sid[m]; okid[i] = sok[m];
  }

  const float* Ws[3] = { WkL + (size_t)t * DINN * HIDD,
                         WqL + (size_t)t * DINN * HIDD,
                         WvL + (size_t)t * DINN * HIDD };
  float* outs[3] = { kb, qb, vb };

  for (int mat = 0; mat < 3; ++mat) {
    const float* W = Ws[mat];
    v8f c = {};
    for (int kk = 0; kk < DINN / 4; ++kk) {
      int k0 = kk * 4 + koff;
      __builtin_prefetch(&W[(kk * 4 + 4) * HIDD + wt * 16], 0, 1);
      v2f a; a.x = sA[rc * LDA + k0];           a.y = sA[rc * LDA + k0 + 1];
      v2f b; b.x = W[k0 * HIDD + wt * 16 + rc]; b.y = W[(k0 + 1) * HIDD + wt * 16 + rc];
      c = wmma_f32_k4(a, b, c);
    }
    float* out = outs[mat];
    if (full) {
#pragma unroll
      for (int i = 0; i < 8; ++i)
        out[(size_t)rowid[i] * HIDD + wt * 16 + rc] = c[i];
    } else {
#pragma unroll
      for (int i = 0; i < 8; ++i)
        if (okid[i] >= 0) out[(size_t)okid[i] * HIDD + wt * 16 + rc] = c[i];
    }
  }
}

// ------------------- typed single-matrix GEMM (Wa) -------------------------

__global__ __launch_bounds__(256) void typed_gemm_kernel(
    const float* __restrict__ xin, const float* __restrict__ WL,
    const int* __restrict__ cnt_nt, const int* __restrict__ nodelist,
    float* __restrict__ out) {
  const int t = blockIdx.y;
  const int cnt = cnt_nt[t];
  const int tile = blockIdx.x;
  if (tile * 16 >= cnt) return;

  __shared__ int sid[16];
  __shared__ int sok[16];
  __shared__ __align__(16) float sA[16 * LDA];
  if (threadIdx.x < 16) {
    int idx = tile * 16 + threadIdx.x;
    int n = (idx < cnt) ? nodelist[t * NN + idx] : -1;
    sok[threadIdx.x] = n;
    sid[threadIdx.x] = (n < 0) ? 0 : n;
  }
  __syncthreads();

  {
    const int tid = threadIdx.x;
#pragma unroll
    for (int it = 0; it < 2; ++it) {
      int qd = it * 256 + tid;
      int row = qd >> 5;
      int c4 = (qd & 31) * 4;
      stage16(xin + (size_t)sid[row] * HIDD + c4, &sA[row * LDA + c4]);
    }
    stage_fence();
  }

  const int lane = threadIdx.x & 31;
  const int wt = threadIdx.x >> 5;
  const int rc = lane & 15;
  const int koff = (lane < 16) ? 0 : 2;
  const bool full = (tile * 16 + 16) <= cnt;

  int rowid[8]; int okid[8];
#pragma unroll
  for (int i = 0; i < 8; ++i) {
    int m = i + ((lane < 16) ? 0 : 8);
    rowid[i] = sid[m]; okid[i] = sok[m];
  }

  const float* W = WL + (size_t)t * HIDD * HIDD;
  v8f c = {};
  for (int kk = 0; kk < HIDD / 4; ++kk) {
    int k0 = kk * 4 + koff;
    __builtin_prefetch(&W[(kk * 4 + 4) * HIDD + wt * 16], 0, 1);
    v2f a; a.x = sA[rc * LDA + k0];           a.y = sA[rc * LDA + k0 + 1];
    v2f b; b.x = W[k0 * HIDD + wt * 16 + rc]; b.y = W[(k0 + 1) * HIDD + wt * 16 + rc];
    c = wmma_f32_k4(a, b, c);
  }
  if (full) {
#pragma unroll
    for (int i = 0; i < 8; ++i)
      out[(size_t)rowid[i] * HIDD + wt * 16 + rc] = c[i];
  } else {
#pragma unroll
    for (int i = 0; i < 8; ++i)
      if (okid[i] >= 0) out[(size_t)okid[i] * HIDD + wt * 16 + rc] = c[i];
  }
}

// ---------------- per-edge attention logits (WMMA + reduce) ----------------
// grid = (EE/16, RT), block = 256 (wave w = head w). 16 edges/tile.

__global__ __launch_bounds__(256) void edge_attn_kernel(
    const float* __restrict__ kb, const float* __restrict__ qb,
    const int* __restrict__ src, const int* __restrict__ dst,
    const float* __restrict__ WattL, const float* __restrict__ priL,
    const int* __restrict__ cnt_rt, const int* __restrict__ edgelist,
    float* __restrict__ abuf, float* __restrict__ amax) {
  const int t = blockIdx.y;
  const int cnt = cnt_rt[t];
  const int tile = blockIdx.x;
  if (tile * 16 >= cnt) return;

  __shared__ int se[16], ssrc[16], sdst[16];
  if (threadIdx.x < 16) {
    int idx = tile * 16 + threadIdx.x;
    int e = (idx < cnt) ? edgelist[t * EE + idx] : -1;
    se[threadIdx.x] = e;
    ssrc[threadIdx.x] = (e < 0) ? 0 : src[e];
    sdst[threadIdx.x] = (e < 0) ? 0 : dst[e];
  }
  __syncthreads();

  const int lane = threadIdx.x & 31;
  const int h = threadIdx.x >> 5;
  const int rc = lane & 15;
  const int koff = (lane < 16) ? 0 : 2;
  const float* W = WattL + ((size_t)h * RTT + t) * (HS * HS);  // [d_in][d_out]
  const float* krow = kb + (size_t)ssrc[rc] * HIDD + h * HS;

  v8f c = {};
#pragma unroll
  for (int kk = 0; kk < 4; ++kk) {
    int k0 = kk * 4 + koff;
    v2f a; a.x = krow[k0];          a.y = krow[k0 + 1];
    v2f b; b.x = W[k0 * HS + rc];   b.y = W[(k0 + 1) * HS + rc];
    c = wmma_f32_k4(a, b, c);
  }

  const float prih = priL[h * RTT + t] * 0.25f;   // 1/sqrt(16) = 0.25
#pragma unroll
  for (int i = 0; i < 8; ++i) {
    int m = i + ((lane < 16) ? 0 : 8);            // edge row of this D slot
    // kw[m][rc] * q[dst[m]][h][rc], reduced over rc within 16-lane halves
    float p = c[i] * qb[(size_t)sdst[m] * HIDD + h * HS + rc];
    p += __shfl_xor(p, 1, 16);
    p += __shfl_xor(p, 2, 16);
    p += __shfl_xor(p, 4, 16);
    p += __shfl_xor(p, 8, 16);
    if (rc == 0 && se[m] >= 0) {
      float av = p * prih;
      abuf[(size_t)se[m] * HH + h] = av;
      atomicMaxF(&amax[(size_t)sdst[m] * HH + h], av);
    }
  }
}

// ------------------------- edge softmax (exp + sum) ------------------------

__global__ void softmax_kernel(const int* __restrict__ dst, float* __restrict__ abuf,
                               const float* __restrict__ amax, float* __restrict__ asum) {
  int i = blockIdx.x * blockDim.x + threadIdx.x;
  if (i >= EE * HH) return;
  int e = i >> 3, h = i & 7;
  int d = dst[e];
  float ex = expf(abuf[i] - amax[d * HH + h]);
  abuf[i] = ex;
  atomicAdd(&asum[d * HH + h], ex);
}

// --------------- per-edge message transform + weighted scatter -------------

__global__ __launch_bounds__(256) void edge_msg_kernel(
    const float* __restrict__ vb,
    const int* __restrict__ src, const int* __restrict__ dst,
    const float* __restrict__ WmsgL,
    const int* __restrict__ cnt_rt, const int* __restrict__ edgelist,
    const float* __restrict__ abuf, const float* __restrict__ asum,
    float* __restrict__ hag) {
  const int t = blockIdx.y;
  const int cnt = cnt_rt[t];
  const int tile = blockIdx.x;
  if (tile * 16 >= cnt) return;

  __shared__ int se[16], ssrc[16], sdst[16];
  if (threadIdx.x < 16) {
    int idx = tile * 16 + threadIdx.x;
    int e = (idx < cnt) ? edgelist[t * EE + idx] : -1;
    se[threadIdx.x] = e;
    ssrc[threadIdx.x] = (e < 0) ? 0 : src[e];
    sdst[threadIdx.x] = (e < 0) ? 0 : dst[e];
  }
  __syncthreads();

  const int lane = threadIdx.x & 31;
  const int h = threadIdx.x >> 5;
  const int rc = lane & 15;
  const int koff = (lane < 16) ? 0 : 2;
  const float* W = WmsgL + ((size_t)h * RTT + t) * (HS * HS);
  const float* vrow = vb + (size_t)ssrc[rc] * HIDD + h * HS;

  v8f c = {};
#pragma unroll
  for (int kk = 0; kk < 4; ++kk) {
    int k0 = kk * 4 + koff;
    v2f a; a.x = vrow[k0];          a.y = vrow[k0 + 1];
    v2f b; b.x = W[k0 * HS + rc];   b.y = W[(k0 + 1) * HS + rc];
    c = wmma_f32_k4(a, b, c);
  }

#pragma unroll
  for (int i = 0; i < 8; ++i) {
    int m = i + ((lane < 16) ? 0 : 8);
    if (se[m] >= 0) {
      float att = abuf[(size_t)se[m] * HH + h] / asum[(size_t)sdst[m] * HH + h];
      atomicAdd(&hag[(size_t)sdst[m] * HIDD + h * HS + rc], c[i] * att);
    }
  }
}

// --------------- SiLU + typed skip-gate + LayerNorm (wave/node) ------------

__global__ __launch_bounds__(256) void node_post_kernel(
    const float* __restrict__ tmp, const float* __restrict__ hin,
    const int* __restrict__ ntype, const float* __restrict__ skipL,
    const float* __restrict__ g, const float* __restrict__ b,
    float* __restrict__ hout) {
  int n = blockIdx.x * 8 + (threadIdx.x >> 5);
  if (n >= NN) return;
  int lane = threadIdx.x & 31;
  float alpha = 1.0f / (1.0f + expf(-skipL[ntype[n]]));
  float vals[4];
  float s = 0.0f, s2 = 0.0f;
#pragma unroll
  for (int j = 0; j < 4; ++j) {
    int cidx = lane + 32 * j;
    float raw = tmp[(size_t)n * HIDD + cidx];
    float sil = raw / (1.0f + expf(-raw));
    float mix = sil * alpha + hin[(size_t)n * HIDD + cidx] * (1.0f - alpha);
    vals[j] = mix; s += mix; s2 += mix * mix;
  }
#pragma unroll
  for (int o = 16; o > 0; o >>= 1) {
    s  += __shfl_xor(s, o, 32);
    s2 += __shfl_xor(s2, o, 32);
  }
  float mu = s * (1.0f / HIDD);
  float var = s2 * (1.0f / HIDD) - mu * mu;
  float inv = rsqrtf(var + 1e-5f);
#pragma unroll
  for (int j = 0; j < 4; ++j) {
    int cidx = lane + 32 * j;
    hout[(size_t)n * HIDD + cidx] = (vals[j] - mu) * inv * g[cidx] + b[cidx];
  }
}

// ------------------------- final FF GEMM (untyped) -------------------------

__global__ __launch_bounds__(256) void ff_kernel(
    const float* __restrict__ hin, const float* __restrict__ ffw,
    const float* __restrict__ ffb, float* __restrict__ out) {
  const int tile = blockIdx.x;
  __shared__ __align__(16) float sA[16 * LDA];
  {
    const int tid = threadIdx.x;
#pragma unroll
    for (int it = 0; it < 2; ++it) {
      int qd = it * 256 + tid;
      int row = qd >> 5;
      int c4 = (qd & 31) * 4;
      stage16(hin + (size_t)(tile * 16 + row) * HIDD + c4, &sA[row * LDA + c4]);
    }
    stage_fence();
  }

  const int lane = threadIdx.x & 31;
  const int wt = threadIdx.x >> 5;
  const int rc = lane & 15;
  const int koff = (lane < 16) ? 0 : 2;

  v8f c = {};
  for (int kk = 0; kk < HIDD / 4; ++kk) {
    int k0 = kk * 4 + koff;
    __builtin_prefetch(&ffw[(kk * 4 + 4) * OUTD + wt * 16], 0, 1);
    v2f a; a.x = sA[rc * LDA + k0];             a.y = sA[rc * LDA + k0 + 1];
    v2f b; b.x = ffw[k0 * OUTD + wt * 16 + rc]; b.y = ffw[(k0 + 1) * OUTD + wt * 16 + rc];
    c = wmma_f32_k4(a, b, c);
  }
  float bias = ffb[wt * 16 + rc];
#pragma unroll
  for (int i = 0; i < 8; ++i) {
    int m = i + ((lane < 16) ? 0 : 8);
    out[(size_t)(tile * 16 + m) * OUTD + wt * 16 + rc] = c[i] + bias;
  }
}

// ------------------------------- launcher ----------------------------------

extern "C" void kernel_launch(void* const* d_in, const int* in_sizes, int n_in,
                              void* d_out, int out_size, void* d_ws, size_t ws_size,
                              hipStream_t stream) {
  const float* x     = (const float*)d_in[0];
  const int*   ntype = (const int*)d_in[1];
  const int*   etype = (const int*)d_in[2];
  const int*   src   = (const int*)d_in[3];
  const int*   dst   = (const int*)d_in[4];
  const float* Wk    = (const float*)d_in[5];
  const float* Wq    = (const float*)d_in[6];
  const float* Wv    = (const float*)d_in[7];
  const float* Wa    = (const float*)d_in[8];
  const float* Watt  = (const float*)d_in[9];
  const float* Wmsg  = (const float*)d_in[10];
  const float* pri   = (const float*)d_in[11];
  const float* skip  = (const float*)d_in[12];
  const float* ln_g  = (const float*)d_in[13];
  const float* ln_b  = (const float*)d_in[14];
  const float* ffw   = (const float*)d_in[15];
  const float* ffb   = (const float*)d_in[16];
  float* outp = (float*)d_out;

  // workspace carve-up (256B aligned slabs)
  char* ws = (char*)d_ws;
  size_t off = 0;
  auto carve = [&](size_t bytes) -> void* {
    void* p = ws + off;
    off += (bytes + 255) & ~(size_t)255;
    return p;
  };
  float* hbuf0 = (float*)carve((size_t)NN * HIDD * 4);
  float* hbuf1 = (float*)carve((size_t)NN * HIDD * 4);
  float* kb    = (float*)carve((size_t)NN * HIDD * 4);   // also reused as Wa output
  float* qb    = (float*)carve((size_t)NN * HIDD * 4);
  float* vb    = (float*)carve((size_t)NN * HIDD * 4);
  float* hag   = (float*)carve((size_t)NN * HIDD * 4);
  float* abuf  = (float*)carve((size_t)EE * HH * 4);
  float* amax  = (float*)carve((size_t)NN * HH * 4);
  float* asum  = (float*)carve((size_t)NN * HH * 4);
  int* cnt_nt   = (int*)carve(NTT * 4);
  int* cnt_rt   = (int*)carve(RTT * 4);
  int* nodelist = (int*)carve((size_t)NTT * NN * 4);
  int* edgelist = (int*)carve((size_t)RTT * EE * 4);

  // bucket once (deterministic per call)
  init_counts_kernel<<<1, 64, 0, stream>>>(cnt_nt, cnt_rt);
  bucket_nodes_kernel<<<(NN + 255) / 256, 256, 0, stream>>>(ntype, cnt_nt, nodelist);
  bucket_edges_kernel<<<(EE + 255) / 256, 256, 0, stream>>>(etype, cnt_rt, edgelist);

  for (int l = 0; l < LLL; ++l) {
    const float* hin = (l == 0) ? x : hbuf0;
    float* hout = (l == 0) ? hbuf0 : hbuf1;
    const size_t wOff = (size_t)l * NTT * DINN * HIDD;
    const size_t aOff = (size_t)l * HH * RTT * HS * HS;

    reset_kernel<<<(NN * HIDD + 255) / 256, 256, 0, stream>>>(hag, amax, asum);
    kqv_kernel<<<dim3(NN / 16, NTT), 256, 0, stream>>>(
        hin, Wk + wOff, Wq + wOff, Wv + wOff, cnt_nt, nodelist, kb, qb, vb);
    edge_attn_kernel<<<dim3(EE / 16, RTT), 256, 0, stream>>>(
        kb, qb, src, dst, Watt + aOff, pri + (size_t)l * HH * RTT,
        cnt_rt, edgelist, abuf, amax);
    softmax_kernel<<<(EE * HH + 255) / 256, 256, 0, stream>>>(dst, abuf, amax, asum);
    edge_msg_kernel<<<dim3(EE / 16, RTT), 256, 0, stream>>>(
        vb, src, dst, Wmsg + aOff, cnt_rt, edgelist, abuf, asum, hag);
    typed_gemm_kernel<<<dim3(NN / 16, NTT), 256, 0, stream>>>(
        hag, Wa + wOff, cnt_nt, nodelist, kb /* tmp */);
    node_post_kernel<<<(NN + 7) / 8, 256, 0, stream>>>(
        kb, hin, ntype, skip + (size_t)l * NTT,
        ln_g + (size_t)l * HIDD, ln_b + (size_t)l * HIDD, hout);
  }

  ff_kernel<<<NN / 16, 256, 0, stream>>>(hbuf1, ffw, ffb, outp);
}